// SNNCore_22256520528169
// MI455X (gfx1250) — compile-verified
//
#include <hip/hip_runtime.h>
#include <stdint.h>

typedef float v2f __attribute__((ext_vector_type(2)));
typedef float v8f __attribute__((ext_vector_type(8)));

#define BM 128
#define BN 64
#define BK 16
// LDS row stride in floats: 80B rows are 16B-aligned for async B128 writes, and
// 20*r mod 64 is distinct for r=0..15 -> conflict-free WMMA fragment reads.
#define LDSK 20

__device__ __forceinline__ uint32_t lds_off(const void* p) {
  // low 32 bits of a flat shared-aperture pointer are the LDS byte address
  return (uint32_t)(uintptr_t)p;
}
__device__ __forceinline__ void async_ld_b128(uint32_t lds, const float* g) {
  asm volatile("global_load_async_to_lds_b128 %0, %1, off" :: "v"(lds), "v"(g) : "memory");
}
__device__ __forceinline__ void async_ld_b32(uint32_t lds, const float* g) {
  asm volatile("global_load_async_to_lds_b32 %0, %1, off" :: "v"(lds), "v"(g) : "memory");
}
__device__ __forceinline__ void wait_async0() {
  asm volatile("s_wait_asynccnt 0x0" ::: "memory");
}

// C[M,N] = A[M,K] @ B + optional per-column bias
// BT=true : B is [N,K] row-major -> C = A * B^T   (contiguous-K staging, B128)
// BT=false: B is [K,N] row-major -> C = A * B     (transpose scatter, B32)
// Requires: M%BM==0, N%BN==0, (K/BK) even.
template <bool BT>
__global__ __launch_bounds__(128) void gemm_f32_wmma(
    const float* __restrict__ A, const float* __restrict__ B,
    const float* __restrict__ bias, float* __restrict__ C,
    int M, int N, int K, int lda, int ldb, int ldc) {
  __shared__ float sA[2][BM * LDSK];
  __shared__ float sB[2][BN * LDSK];  // stored [n][k]

  const int tid = threadIdx.x;
  const int lane = tid & 31;
  const int wave = tid >> 5;      // 0..3
  const int wm = wave >> 1;       // 0..1 : 64-row slab
  const int wn = wave & 1;        // 0..1 : 32-col slab
  const int lrow = lane & 15;
  const int half = lane >> 4;     // lane half selects K pair / M+8
  const int blockM = blockIdx.y * BM;
  const int blockN = blockIdx.x * BN;

  // ---- per-thread staging pointers (computed once; advanced per K-chunk) ----
  const float* aG[4];
  uint32_t aL0[4], aL1[4];
#pragma unroll
  for (int i = 0; i < 4; ++i) {
    int id = i * 128 + tid;
    int r = id >> 2, c4 = id & 3;               // 128 rows x 4 float4-groups
    aG[i] = A + (size_t)(blockM + r) * lda + c4 * 4;
    aL0[i] = lds_off(&sA[0][r * LDSK + c4 * 4]);
    aL1[i] = lds_off(&sA[1][r * LDSK + c4 * 4]);
  }
  const float* bG[8];
  uint32_t bL0[8], bL1[8];
  if (BT) {
#pragma unroll
    for (int i = 0; i < 2; ++i) {
      int id = i * 128 + tid;
      int n = id >> 2, c4 = id & 3;             // 64 rows x 4 float4-groups
      bG[i] = B + (size_t)(blockN + n) * ldb + c4 * 4;
      bL0[i] = lds_off(&sB[0][n * LDSK + c4 * 4]);
      bL1[i] = lds_off(&sB[1][n * LDSK + c4 * 4]);
    }
  } else {
#pragma unroll
    for (int i = 0; i < 8; ++i) {
      int id = i * 128 + tid;
      int k = id >> 6, n = id & 63;             // 16 k-rows x 64 n (coalesced in n)
      bG[i] = B + (size_t)k * ldb + blockN + n;
      bL0[i] = lds_off(&sB[0][n * LDSK + k]);   // transpose scatter into [n][k]
      bL1[i] = lds_off(&sB[1][n * LDSK + k]);
    }
  }

  // compile-time buffer select -> no indexed VGPR moves in the hot loop
  auto stage0 = [&]() {
#pragma unroll
    for (int i = 0; i < 4; ++i) { async_ld_b128(aL0[i], aG[i]); aG[i] += BK; }
    if (BT) {
#pragma unroll
      for (int i = 0; i < 2; ++i) { async_ld_b128(bL0[i], bG[i]); bG[i] += BK; }
    } else {
#pragma unroll
      for (int i = 0; i < 8; ++i) { async_ld_b32(bL0[i], bG[i]); bG[i] += (size_t)BK * ldb; }
    }
  };
  auto stage1 = [&]() {
#pragma unroll
    for (int i = 0; i < 4; ++i) { async_ld_b128(aL1[i], aG[i]); aG[i] += BK; }
    if (BT) {
#pragma unroll
      for (int i = 0; i < 2; ++i) { async_ld_b128(bL1[i], bG[i]); bG[i] += BK; }
    } else {
#pragma unroll
      for (int i = 0; i < 8; ++i) { async_ld_b32(bL1[i], bG[i]); bG[i] += (size_t)BK * ldb; }
    }
  };

  v8f acc[4][2] = {};

  auto compute = [&](const float* __restrict__ la, const float* __restrict__ lb) {
#pragma unroll
    for (int k4 = 0; k4 < BK / 4; ++k4) {
      const int kk = k4 * 4 + 2 * half;  // lanes 16-31 hold K+2 (ISA A/B layout)
      v2f a[4], b[2];
#pragma unroll
      for (int mi = 0; mi < 4; ++mi) {
        int row = wm * 64 + mi * 16 + lrow;
        a[mi].x = la[row * LDSK + kk];
        a[mi].y = la[row * LDSK + kk + 1];
      }
#pragma unroll
      for (int ni = 0; ni < 2; ++ni) {
        int col = wn * 32 + ni * 16 + lrow;
        b[ni].x = lb[col * LDSK + kk];
        b[ni].y = lb[col * LDSK + kk + 1];
      }
#pragma unroll
      for (int mi = 0; mi < 4; ++mi)
#pragma unroll
        for (int ni = 0; ni < 2; ++ni)
          acc[mi][ni] = __builtin_amdgcn_wmma_f32_16x16x4_f32(
              false, a[mi], false, b[ni], (short)0, acc[mi][ni], false, false);
    }
  };

  const int nChunks = K / BK;  // even by contract

  stage0();
  wait_async0();
  __syncthreads();

  for (int ch = 0; ch < nChunks; ch += 2) {
    stage1();                       // prefetch chunk ch+1 into buffer 1
    compute(sA[0], sB[0]);          // consume buffer 0
    wait_async0();
    __syncthreads();
    if (ch + 2 < nChunks) stage0(); // prefetch chunk ch+2 into buffer 0
    compute(sA[1], sB[1]);          // consume buffer 1
    wait_async0();
    __syncthreads();
  }

  // ---- epilogue: C layout VGPR r -> M = r (+8 for lane half 1), N = lane ----
#pragma unroll
  for (int mi = 0; mi < 4; ++mi) {
#pragma unroll
    for (int ni = 0; ni < 2; ++ni) {
      int col = blockN + wn * 32 + ni * 16 + lrow;
      float bv = bias ? bias[col] : 0.0f;
#pragma unroll
      for (int r = 0; r < 8; ++r) {
        int row = blockM + wm * 64 + mi * 16 + r + 8 * half;
        C[(size_t)row * ldc + col] = acc[mi][ni][r] + bv;
      }
    }
  }
}

// bc[g] = dot(W_hid[g,:], b_proj) + b_hid[g]
__global__ void combine_bias(const float* __restrict__ Wh,
                             const float* __restrict__ bp,
                             const float* __restrict__ bh,
                             float* __restrict__ bc, int H) {
  int g = blockIdx.x * blockDim.x + threadIdx.x;
  if (g >= H) return;
  const float* row = Wh + (size_t)g * H;
  float s = 0.0f;
  for (int h = 0; h < H; ++h) s += row[h] * bp[h];
  bc[g] = s + bh[g];
}

// LIF over T (tau=2, V_TH=1, soft reset) + time-mean, 4 hidden units per thread.
__global__ void lif_mean(const float* __restrict__ h2, float* __restrict__ pooled,
                         int Bsz, int T, int H) {
  const int H4 = H >> 2;
  int idx = blockIdx.x * blockDim.x + threadIdx.x;
  if (idx >= Bsz * H4) return;
  int b = idx / H4, g4 = idx % H4;
  const float4* p = (const float4*)(h2 + (size_t)b * T * H) + g4;
  float v0 = 0.f, v1 = 0.f, v2 = 0.f, v3 = 0.f;
  float c0 = 0.f, c1 = 0.f, c2 = 0.f, c3 = 0.f;
#pragma unroll 4
  for (int t = 0; t < T; ++t) {
    float4 xt = p[(size_t)t * H4];
    v0 = 0.5f * (v0 + xt.x); { float s = (v0 >= 1.f) ? 1.f : 0.f; c0 += s; v0 -= s; }
    v1 = 0.5f * (v1 + xt.y); { float s = (v1 >= 1.f) ? 1.f : 0.f; c1 += s; v1 -= s; }
    v2 = 0.5f * (v2 + xt.z); { float s = (v2 >= 1.f) ? 1.f : 0.f; c2 += s; v2 -= s; }
    v3 = 0.5f * (v3 + xt.w); { float s = (v3 >= 1.f) ? 1.f : 0.f; c3 += s; v3 -= s; }
  }
  const float inv = 1.0f / (float)T;
  float4 r; r.x = c0 * inv; r.y = c1 * inv; r.z = c2 * inv; r.w = c3 * inv;
  ((float4*)pooled)[idx] = r;
}

// out[b,o] = dot(pooled[b,:], W_out[o,:]) + b_out[o]
__global__ void head_kernel(const float* __restrict__ pooled,
                            const float* __restrict__ Wo,
                            const float* __restrict__ bo,
                            float* __restrict__ out, int Bsz, int H, int O) {
  int tid = blockIdx.x * blockDim.x + threadIdx.x;
  if (tid >= Bsz * O) return;
  int b = tid / O, o = tid % O;
  const float* pv = pooled + (size_t)b * H;
  const float* wv = Wo + (size_t)o * H;
  float s = 0.0f;
  for (int h = 0; h < H; ++h) s += pv[h] * wv[h];
  out[tid] = s + bo[o];
}

extern "C" void kernel_launch(void* const* d_in, const int* in_sizes, int n_in,
                              void* d_out, int out_size, void* d_ws, size_t ws_size,
                              hipStream_t stream) {
  const float* x  = (const float*)d_in[0];  // [32,1024,1024]
  const float* Wp = (const float*)d_in[1];  // [1024,1024]
  const float* bp = (const float*)d_in[2];  // [1024]
  const float* Wh = (const float*)d_in[3];  // [1024,1024]
  const float* bh = (const float*)d_in[4];  // [1024]
  const float* Wo = (const float*)d_in[5];  // [10,1024]
  const float* bo = (const float*)d_in[6];  // [10]
  float* out = (float*)d_out;               // [32,10]

  const int Bsz = 32, T = 1024, D = 1024, H = 1024, O = 10;
  const int NR = Bsz * T;  // 32768 token rows

  // workspace layout
  char* ws = (char*)d_ws;
  float* Wc = (float*)ws;                                   // [H,D]   4 MB
  float* bc = (float*)(ws + (size_t)H * D * 4);             // [H]     4 KB
  float* pooled = (float*)(ws + (size_t)H * D * 4 + 4096);  // [B,H] 128 KB
  float* h2 = (float*)(ws + (size_t)H * D * 4 + 4096 + (size_t)Bsz * H * 4);  // [NR,H] 128 MB

  // 1) combined bias: bc = Wh @ bp + bh
  combine_bias<<<H / 256, 256, 0, stream>>>(Wh, bp, bh, bc, H);
  // 2) Wc = W_hid @ W_proj  (NN: B=[K,N])
  gemm_f32_wmma<false><<<dim3(D / BN, H / BM), 128, 0, stream>>>(
      Wh, Wp, nullptr, Wc, H, D, H, H, D, D);
  // 3) h2 = x @ Wc^T + bc   (NT: B=[N,K])
  gemm_f32_wmma<true><<<dim3(H / BN, NR / BM), 128, 0, stream>>>(
      x, Wc, bc, h2, NR, H, D, D, D, H);
  // 4) LIF scan + time-mean of spikes (float4 per thread)
  lif_mean<<<(Bsz * (H / 4)) / 256, 256, 0, stream>>>(h2, pooled, Bsz, T, H);
  // 5) classification head
  head_kernel<<<1, 320, 0, stream>>>(pooled, Wo, bo, out, Bsz, H, O);
}